// yoloLoss_43593918054773
// MI455X (gfx1250) — compile-verified
//
#include <hip/hip_runtime.h>

// ---------------------------------------------------------------------------
// YOLO-v1 loss, MI455X (gfx1250, wave32).
// Memory-bound streaming reduction: 193 MB in, 4 B out -> ~8.3 us floor at
// 23.3 TB/s. Strategy: NT float2 loads, per-thread cell loss, wave reduction
// via V_WMMA_F32_16X16X4_F32 (A = partials, B = ones), LDS across 8 waves,
// deterministic two-stage block reduction (no FP atomics).
// ---------------------------------------------------------------------------

typedef __attribute__((ext_vector_type(2))) float v2f;
typedef __attribute__((ext_vector_type(8))) float v8f;

#define N_CH      30
#define INV_S     (1.0f / 14.0f)
#define L_COORD   5.0f
#define L_NOOBJ   0.5f
#define BLOCK     256
#define WAVES_PER_BLOCK (BLOCK / 32)

// Wave-level f32 sum using the FP32 WMMA path.
// A (16x4 f32): vgpr0 = partial (K=0 lanes 0-15, K=2 lanes 16-31), vgpr1 = 0.
// B (4x16 f32): all ones.  D[m][n] = partial[m] + partial[m+16] for all n.
// Per-lane sum of the 8 C/D VGPRs gives the half-wave row sums; one
// shfl_xor(16) combines the two halves. Requires EXEC == all 1s (guaranteed:
// called only at full reconvergence).
__device__ __forceinline__ float wave_reduce_wmma(float x) {
    v2f a; a[0] = x;    a[1] = 0.0f;
    v2f b; b[0] = 1.0f; b[1] = 1.0f;
    v8f c = {};
    c = __builtin_amdgcn_wmma_f32_16x16x4_f32(
        /*neg_a=*/false, a, /*neg_b=*/false, b,
        /*c_mod=*/(short)0, c, /*reuse_a=*/false, /*reuse_b=*/false);
    float s = c[0] + c[1] + c[2] + c[3] + c[4] + c[5] + c[6] + c[7];
    s += __shfl_xor(s, 16, 32);   // combine M=0..7 half with M=8..15 half
    return s;                     // every lane holds the wave total
}

__device__ __forceinline__ float block_reduce(float x, float* smem) {
    const int lane = threadIdx.x & 31;
    const int wid  = threadIdx.x >> 5;
    float ws = wave_reduce_wmma(x);
    if (lane == 0) smem[wid] = ws;
    __syncthreads();
    float total = 0.0f;
    if (threadIdx.x == 0) {
#pragma unroll
        for (int i = 0; i < WAVES_PER_BLOCK; ++i) total += smem[i];
    }
    return total;  // valid on thread 0 only
}

__global__ void __launch_bounds__(BLOCK)
yolo_loss_stage1(const float* __restrict__ pred,
                 const float* __restrict__ targ,
                 float* __restrict__ partials,
                 int nCells) {
    const int cell = blockIdx.x * BLOCK + threadIdx.x;
    float loss = 0.0f;

    if (cell < nCells) {
        // 30 floats per tensor per cell; base offset cell*120B is 8B-aligned
        // -> 15 x global_load_b64 each, non-temporal (streamed once).
        const v2f* p2 = reinterpret_cast<const v2f*>(pred + (size_t)cell * N_CH);
        const v2f* t2 = reinterpret_cast<const v2f*>(targ + (size_t)cell * N_CH);
        float p[N_CH], t[N_CH];
#pragma unroll
        for (int i = 0; i < N_CH / 2; ++i) {
            v2f v = __builtin_nontemporal_load(p2 + i);
            p[2 * i] = v[0]; p[2 * i + 1] = v[1];
        }
#pragma unroll
        for (int i = 0; i < N_CH / 2; ++i) {
            v2f v = __builtin_nontemporal_load(t2 + i);
            t[2 * i] = v[0]; t[2 * i + 1] = v[1];
        }

        const float m  = (t[4] > 0.0f) ? 1.0f : 0.0f;
        const float nm = 1.0f - m;

        // Target box 0 in xyxy (reference: xy/S +- 0.5*wh).
        const float tx1 = t[0] * INV_S - 0.5f * t[2];
        const float ty1 = t[1] * INV_S - 0.5f * t[3];
        const float tx2 = t[0] * INV_S + 0.5f * t[2];
        const float ty2 = t[1] * INV_S + 0.5f * t[3];
        const float ta  = (tx2 - tx1) * (ty2 - ty1);

        float iou0, iou1;
#pragma unroll
        for (int k = 0; k < 2; ++k) {
            const int bofs = 5 * k;
            const float px1 = p[bofs + 0] * INV_S - 0.5f * p[bofs + 2];
            const float py1 = p[bofs + 1] * INV_S - 0.5f * p[bofs + 3];
            const float px2 = p[bofs + 0] * INV_S + 0.5f * p[bofs + 2];
            const float py2 = p[bofs + 1] * INV_S + 0.5f * p[bofs + 3];
            const float pa  = (px2 - px1) * (py2 - py1);
            const float iw  = fmaxf(fminf(px2, tx2) - fmaxf(px1, tx1), 0.0f);
            const float ih  = fmaxf(fminf(py2, ty2) - fmaxf(py1, ty1), 0.0f);
            const float inter = iw * ih;
            const float v = inter / (pa + ta - inter);
            if (k == 0) iou0 = v; else iou1 = v;
        }

        // argmax (first max on ties, matching jnp.argmax).
        const bool  sel     = (iou1 > iou0);
        const float max_iou = sel ? iou1 : iou0;
        const float pr0 = sel ? p[5] : p[0];
        const float pr1 = sel ? p[6] : p[1];
        const float pr2 = sel ? p[7] : p[2];
        const float pr3 = sel ? p[8] : p[3];
        const float pr4 = sel ? p[9] : p[4];
        const float tr0 = sel ? t[5] : t[0];
        const float tr1 = sel ? t[6] : t[1];
        const float tr2 = sel ? t[7] : t[2];
        const float tr3 = sel ? t[8] : t[3];

        const float dx = pr0 - tr0, dy = pr1 - tr1;
        const float loss_xy = dx * dx + dy * dy;

        const float dw = sqrtf(pr2) - sqrtf(tr2);
        const float dh = sqrtf(pr3) - sqrtf(tr3);
        const float loss_wh = dw * dw + dh * dh;

        const float dob = pr4 - max_iou;
        const float loss_obj = dob * dob;

        float loss_class = 0.0f;
#pragma unroll
        for (int c = 0; c < 20; ++c) {
            const float d = p[10 + c] - t[10 + c];
            loss_class = fmaf(d, d, loss_class);
        }

        const float dn0 = p[4] - t[4];
        const float dn1 = p[9] - t[9];
        const float loss_noobj = dn0 * dn0 + dn1 * dn1;

        loss = m * (L_COORD * (loss_xy + loss_wh) + loss_obj + loss_class)
             + L_NOOBJ * nm * loss_noobj;
    }

    // Full reconvergence here -> EXEC all 1s for the WMMA reduction.
    __shared__ float smem[WAVES_PER_BLOCK];
    const float total = block_reduce(loss, smem);
    if (threadIdx.x == 0) partials[blockIdx.x] = total;
}

__global__ void __launch_bounds__(BLOCK)
yolo_loss_stage2(const float* __restrict__ partials, int n,
                 float* __restrict__ out, float invBatch) {
    float s = 0.0f;
    for (int i = threadIdx.x; i < n; i += BLOCK) s += partials[i];
    __shared__ float smem[WAVES_PER_BLOCK];
    const float total = block_reduce(s, smem);
    if (threadIdx.x == 0) out[0] = total * invBatch;
}

extern "C" void kernel_launch(void* const* d_in, const int* in_sizes, int n_in,
                              void* d_out, int out_size, void* d_ws, size_t ws_size,
                              hipStream_t stream) {
    const float* pred = (const float*)d_in[0];
    const float* targ = (const float*)d_in[1];

    const int nElem  = in_sizes[0];            // batch*S*S*N_CH
    const int nCells = nElem / N_CH;           // batch*S*S = 802816
    const int batch  = nCells / (14 * 14);     // 4096
    const int blocks = (nCells + BLOCK - 1) / BLOCK;  // 3136

    float* partials = (float*)d_ws;

    yolo_loss_stage1<<<blocks, BLOCK, 0, stream>>>(pred, targ, partials, nCells);
    yolo_loss_stage2<<<1, BLOCK, 0, stream>>>(partials, blocks, (float*)d_out,
                                              1.0f / (float)batch);
}